// SpectrogramAdaptiveMoving_75505525064576
// MI455X (gfx1250) — compile-verified
//
#include <hip/hip_runtime.h>
#include <cstddef>

typedef float v2f __attribute__((ext_vector_type(2)));
typedef float v8f __attribute__((ext_vector_type(8)));

#define N_BANDS   2049
#define TCOLS     8
#define FRAME     4096
#define BUF       1024
#define PHASE_K   0.00153398078788564f   // 2*pi/4096

#define MSUB      4                      // 4 M-subtiles of 16 -> 64 batch rows / block
#define MTILE     64
#define NWAVES    8                      // 256 threads, each wave owns 16 freq cols
#define KCHUNK    64
#define LDSS      68                     // padded LDS row stride (floats)

// One frame of the multi-resolution spectrogram as a WMMA GEMM:
//   power[b,f] = (sum_k x[b,kb+k]*w(j)*cos(2pi f j/4096))^2
//              + (sum_k x[b,kb+k]*w(j)*sin(2pi f j/4096))^2 ,  j = 4096 - start + (kb+k)
// then out[b,f,t0..t0+tn) = min(out, power).  (Launches are serialized -> no races.)
__global__ __launch_bounds__(256) void frame_dft_min(
    const float* __restrict__ x, float* __restrict__ out,
    int start, int kb, int kcount, int stride, int t0, int tn)
{
    __shared__ float ax[MTILE * LDSS];

    const int tid   = threadIdx.x;
    const int lane  = tid & 31;
    const int wave  = tid >> 5;
    const int mBase = blockIdx.x * MTILE;
    const int fTile = blockIdx.y * (NWAVES * 16) + wave * 16;

    const int nRow  = lane & 15;   // A row (m) and B/C column (n) owned by this lane
    const int kHalf = lane >> 4;   // which k-pair half (A/B) / which M half (C)
    const int f     = fTile + nRow;

    v8f accRe[MSUB] = {};
    v8f accIm[MSUB] = {};

    const float wscale = (float)stride * (1.0f / 2048.0f);  // stride * 0.5/1024

    for (int kk0 = 0; kk0 < kcount; kk0 += KCHUNK) {
        __syncthreads();
        // Stage 64 rows x 64 cols of x into LDS (1024 float4, 4 per thread).
#pragma unroll
        for (int it = 0; it < (MTILE * KCHUNK / 4) / 256; ++it) {
            int idx = tid + it * 256;
            int row = idx >> 4;        // 16 float4 per row
            int c4  = idx & 15;
            const float* gp = x + (size_t)(mBase + row) * BUF + (kb + kk0 + c4 * 4);
            float4 v = *(const float4*)gp;
            *(float4*)&ax[row * LDSS + c4 * 4] = v;
            if (kk0 + KCHUNK < kcount)
                __builtin_prefetch(gp + KCHUNK, 0, 1);   // global_prefetch_b8 next K-chunk
        }
        __syncthreads();

#pragma unroll 4
        for (int kk = 0; kk < KCHUNK; kk += 4) {
            // B fragments (4x16): lane holds k = kk0+kk+2*kHalf+{0,1}, col n = f.
            v2f bc, bs;
#pragma unroll
            for (int v = 0; v < 2; ++v) {
                int kAbs = kb + kk0 + kk + 2 * kHalf + v;
                int j    = FRAME - start + kAbs;
                // w(j) = stride * hann4096[(j*stride) mod 4096] / 1024
                float w  = wscale * (1.0f - __cosf((float)((j * stride) & 4095) * PHASE_K));
                float ph = (float)((f * j) & 4095) * PHASE_K;
                float sn, cs;
                __sincosf(ph, &sn, &cs);
                bc[v] = w * cs;
                bs[v] = w * sn;
            }
#pragma unroll
            for (int s = 0; s < MSUB; ++s) {
                int col = kk + 2 * kHalf;
                v2f a;
                a[0] = ax[(s * 16 + nRow) * LDSS + col];
                a[1] = ax[(s * 16 + nRow) * LDSS + col + 1];
                accRe[s] = __builtin_amdgcn_wmma_f32_16x16x4_f32(
                    false, a, false, bc, (short)0, accRe[s], false, false);
                accIm[s] = __builtin_amdgcn_wmma_f32_16x16x4_f32(
                    false, a, false, bs, (short)0, accIm[s], false, false);
            }
        }
    }

    // Epilogue: power + running min into out[b, f, t0..t0+tn)
    if (f < N_BANDS) {
#pragma unroll
        for (int s = 0; s < MSUB; ++s) {
#pragma unroll
            for (int r = 0; r < 8; ++r) {
                int b    = mBase + s * 16 + r + 8 * kHalf;    // C layout: M = r + 8*(lane>>4)
                float re = accRe[s][r];
                float im = accIm[s][r];
                float p  = re * re + im * im;
                size_t o = ((size_t)b * N_BANDS + f) * TCOLS + t0;
                for (int t = 0; t < tn; ++t)
                    out[o + t] = fminf(out[o + t], p);
            }
        }
    }
}

__global__ void fill_kernel(float* __restrict__ out, int n, float v) {
    int i = blockIdx.x * blockDim.x + threadIdx.x;
    if (i < n) out[i] = v;
}

__global__ void db_kernel(float* __restrict__ out, int n) {
    int i = blockIdx.x * blockDim.x + threadIdx.x;
    if (i < n) {
        float v = fmaxf(out[i], 1e-10f);
        out[i] = 10.0f * log10f(v);
    }
}

extern "C" void kernel_launch(void* const* d_in, const int* in_sizes, int n_in,
                              void* d_out, int out_size, void* d_ws, size_t ws_size,
                              hipStream_t stream) {
    (void)n_in; (void)d_ws; (void)ws_size;
    const float* x = (const float*)d_in[0];
    float* out = (float*)d_out;
    const int B = in_sizes[0] / BUF;     // 1024
    const int n = out_size;              // B * 2049 * 8

    // min-identity init (harness poisons d_out; must re-init every call)
    fill_kernel<<<(n + 255) / 256, 256, 0, stream>>>(out, n, 1.0e6f);

    // 15 frames: (lvl, i) with start = (1024>>lvl)*(i+1), effective K = min(start, 4096>>lvl)
    dim3 grid(B / MTILE, (N_BANDS + NWAVES * 16 - 1) / (NWAVES * 16));  // 16 x 17
    for (int lvl = 0; lvl < 4; ++lvl) {
        int stride = 1 << lvl;
        int L      = FRAME >> lvl;
        int chunk  = BUF >> lvl;
        int nfr    = 1 << lvl;
        int tspan  = TCOLS >> lvl;
        for (int i = 0; i < nfr; ++i) {
            int start = chunk * (i + 1);
            int kc    = start < L ? start : L;   // window is zero before 4096-L
            int kb    = start - kc;
            frame_dft_min<<<grid, 256, 0, stream>>>(x, out, start, kb, kc, stride,
                                                    i * tspan, tspan);
        }
    }

    db_kernel<<<(n + 255) / 256, 256, 0, stream>>>(out, n);
}